// PriorBCE_57458072486126
// MI455X (gfx1250) — compile-verified
//
#include <hip/hip_runtime.h>
#include <math.h>

#define EMB_C 256

typedef __attribute__((ext_vector_type(2))) float v2f;
typedef __attribute__((ext_vector_type(8))) float v8f;

__device__ __forceinline__ float wave_sum32(float v) {
#pragma unroll
  for (int off = 16; off > 0; off >>= 1) v += __shfl_xor(v, off, 32);
  return v;
}

// merge two online-softmax states (m,s) <- (m,s) + (om,os)
__device__ __forceinline__ void merge_ms(float& m, float& s, float om, float os) {
  float nm = fmaxf(m, om);
  s = s * __expf(m - nm) + os * __expf(om - nm);
  m = nm;
}

// online softmax update with a float4 of logits + dot with x
__device__ __forceinline__ void online4(const float4 a, const float4 xv,
                                        float& m, float& s, float& d) {
  float mx = fmaxf(fmaxf(a.x, a.y), fmaxf(a.z, a.w));
  float nm = fmaxf(m, mx);
  s = s * __expf(m - nm) +
      (__expf(a.x - nm) + __expf(a.y - nm)) + (__expf(a.z - nm) + __expf(a.w - nm));
  m = nm;
  d = fmaf(a.x, xv.x, d);
  d = fmaf(a.y, xv.y, d);
  d = fmaf(a.z, xv.z, d);
  d = fmaf(a.w, xv.w, d);
}

// -------- Kernel 1: per-row fused log-softmax-dot for 3 logit streams ------
// grid = B rows, block = 256 (8 waves). One HBM pass over 4 streams.
__global__ __launch_bounds__(256) void bce_rows_kernel(
    const float4* __restrict__ A0,  // recon_x   (BCE_merged)
    const float4* __restrict__ A1,  // logits_rec (BCE_rec)
    const float4* __restrict__ A2,  // logits_text (BCE_text)
    const float4* __restrict__ X,
    float* __restrict__ wsA, int nv /* = N/4 */) {
  const size_t base = (size_t)blockIdx.x * (size_t)nv;
  float m0 = -INFINITY, m1 = -INFINITY, m2 = -INFINITY;
  float s0 = 0.f, s1 = 0.f, s2 = 0.f;
  float d0 = 0.f, d1 = 0.f, d2 = 0.f, sx = 0.f;

  for (int j = threadIdx.x; j < nv; j += 256) {
    float4 xv = X[base + j];
    sx += (xv.x + xv.y) + (xv.z + xv.w);
    online4(A0[base + j], xv, m0, s0, d0);
    online4(A1[base + j], xv, m1, s1, d1);
    online4(A2[base + j], xv, m2, s2, d2);
  }

  // wave32 butterfly reduce
#pragma unroll
  for (int off = 16; off > 0; off >>= 1) {
    merge_ms(m0, s0, __shfl_xor(m0, off, 32), __shfl_xor(s0, off, 32));
    merge_ms(m1, s1, __shfl_xor(m1, off, 32), __shfl_xor(s1, off, 32));
    merge_ms(m2, s2, __shfl_xor(m2, off, 32), __shfl_xor(s2, off, 32));
  }
  d0 = wave_sum32(d0);
  d1 = wave_sum32(d1);
  d2 = wave_sum32(d2);
  sx = wave_sum32(sx);

  __shared__ float sh[8][10];
  const int wave = threadIdx.x >> 5, lane = threadIdx.x & 31;
  if (lane == 0) {
    float* p = sh[wave];
    p[0] = m0; p[1] = s0; p[2] = m1; p[3] = s1; p[4] = m2; p[5] = s2;
    p[6] = d0; p[7] = d1; p[8] = d2; p[9] = sx;
  }
  __syncthreads();
  if (threadIdx.x == 0) {
    float M0 = sh[0][0], S0 = sh[0][1], M1 = sh[0][2], S1 = sh[0][3];
    float M2 = sh[0][4], S2 = sh[0][5];
    float D0 = sh[0][6], D1 = sh[0][7], D2 = sh[0][8], SX = sh[0][9];
#pragma unroll
    for (int w = 1; w < 8; ++w) {
      merge_ms(M0, S0, sh[w][0], sh[w][1]);
      merge_ms(M1, S1, sh[w][2], sh[w][3]);
      merge_ms(M2, S2, sh[w][4], sh[w][5]);
      D0 += sh[w][6]; D1 += sh[w][7]; D2 += sh[w][8]; SX += sh[w][9];
    }
    const float l0 = M0 + __logf(S0);
    const float l1 = M1 + __logf(S1);
    const float l2 = M2 + __logf(S2);
    float* o = wsA + (size_t)blockIdx.x * 3;
    o[0] = D0 - l0 * SX;  // merged contribution:  sum_i logsoftmax*x
    o[1] = D1 - l1 * SX;  // rec
    o[2] = D2 - l2 * SX;  // text
  }
}

// -------- Kernel 2: KLD1/KLD2/Wasserstein integrand sums via WMMA ---------
// Sum-reduction on the matrix unit: D = A * ones(4x16) + C accumulates row
// sums of each 16x4 A chunk into C; sum(C over all 256 entries) = 16*sum(A).
// grid = 64, block = 256; trip count is uniform so EXEC is all-ones (WMMA req).
__global__ __launch_bounds__(256) void kld_wass_kernel(
    const float* __restrict__ mu, const float* __restrict__ lv,
    const float* __restrict__ pmu, const float* __restrict__ plv,
    float* __restrict__ wsB, int total) {
  const int nthreads = gridDim.x * blockDim.x;
  const int gtid = blockIdx.x * blockDim.x + threadIdx.x;
  const int iters = total / (nthreads * 2);  // uniform across all lanes

  v8f c1 = {0.f, 0.f, 0.f, 0.f, 0.f, 0.f, 0.f, 0.f};
  v8f c2 = {0.f, 0.f, 0.f, 0.f, 0.f, 0.f, 0.f, 0.f};
  v8f c3 = {0.f, 0.f, 0.f, 0.f, 0.f, 0.f, 0.f, 0.f};
  const v2f onesB = {1.0f, 1.0f};

  for (int it = 0; it < iters; ++it) {
    const int idx = (it * nthreads + gtid) * 2;  // lane-contiguous pair -> b64 loads
    float u0 = mu[idx],  u1 = mu[idx + 1];
    float v0 = lv[idx],  v1 = lv[idx + 1];
    float q0 = pmu[idx], q1 = pmu[idx + 1];
    float w0 = plv[idx], w1 = plv[idx + 1];

    float e0 = __expf(v0), e1 = __expf(v1);
    float f0 = __expf(w0), f1 = __expf(w1);
    float g0 = __expf(0.5f * (v0 + w0));  // sqrt(e*f), closed form
    float g1 = __expf(0.5f * (v1 + w1));
    float dm0 = u0 - q0, dm1 = u1 - q1;

    v2f a1 = {1.0f + v0 - u0 * u0 - e0, 1.0f + v1 - u1 * u1 - e1};   // KLD1 integrand
    v2f a2 = {1.0f + w0 - q0 * q0 - f0, 1.0f + w1 - q1 * q1 - f1};   // KLD2 integrand
    v2f a3 = {dm0 * dm0 + e0 + f0 - 2.0f * g0,
              dm1 * dm1 + e1 + f1 - 2.0f * g1};                       // BW integrand

    c1 = __builtin_amdgcn_wmma_f32_16x16x4_f32(false, a1, false, onesB,
                                               (short)0, c1, false, false);
    c2 = __builtin_amdgcn_wmma_f32_16x16x4_f32(false, a2, false, onesB,
                                               (short)0, c2, false, false);
    c3 = __builtin_amdgcn_wmma_f32_16x16x4_f32(false, a3, false, onesB,
                                               (short)0, c3, false, false);
  }

  // generic tail (empty for B=4096, EMB=256), plain VALU — divergence is safe here
  float t1 = 0.f, t2 = 0.f, t3 = 0.f;
  for (int idx = iters * nthreads * 2 + gtid; idx < total; idx += nthreads) {
    float u = mu[idx], v = lv[idx], q = pmu[idx], w = plv[idx];
    float e = __expf(v), f = __expf(w), g = __expf(0.5f * (v + w));
    float dm = u - q;
    t1 += 1.0f + v - u * u - e;
    t2 += 1.0f + w - q * q - f;
    t3 += dm * dm + e + f - 2.0f * g;
  }

  float p1 = (((c1[0] + c1[1]) + (c1[2] + c1[3])) + ((c1[4] + c1[5]) + (c1[6] + c1[7]))) * 0.0625f + t1;
  float p2 = (((c2[0] + c2[1]) + (c2[2] + c2[3])) + ((c2[4] + c2[5]) + (c2[6] + c2[7]))) * 0.0625f + t2;
  float p3 = (((c3[0] + c3[1]) + (c3[2] + c3[3])) + ((c3[4] + c3[5]) + (c3[6] + c3[7]))) * 0.0625f + t3;
  p1 = wave_sum32(p1);
  p2 = wave_sum32(p2);
  p3 = wave_sum32(p3);

  __shared__ float sh[8][3];
  const int wave = threadIdx.x >> 5, lane = threadIdx.x & 31;
  if (lane == 0) { sh[wave][0] = p1; sh[wave][1] = p2; sh[wave][2] = p3; }
  __syncthreads();
  if (threadIdx.x == 0) {
    float a = 0.f, b = 0.f, c = 0.f;
#pragma unroll
    for (int w = 0; w < 8; ++w) { a += sh[w][0]; b += sh[w][1]; c += sh[w][2]; }
    float* o = wsB + (size_t)blockIdx.x * 3;
    o[0] = a; o[1] = b; o[2] = c;
  }
}

// -------- Kernel 3: final scalar assembly ---------------------------------
__global__ __launch_bounds__(256) void finalize_kernel(
    const float* __restrict__ wsA, const float* __restrict__ wsB,
    int batch, int n_items, int nB, float* __restrict__ out) {
  float c0 = 0.f, c1 = 0.f, c2 = 0.f, k1 = 0.f, k2 = 0.f, w = 0.f;
  for (int r = threadIdx.x; r < batch; r += 256) {
    c0 += wsA[r * 3 + 0]; c1 += wsA[r * 3 + 1]; c2 += wsA[r * 3 + 2];
  }
  for (int g = threadIdx.x; g < nB; g += 256) {
    k1 += wsB[g * 3 + 0]; k2 += wsB[g * 3 + 1]; w += wsB[g * 3 + 2];
  }
  c0 = wave_sum32(c0); c1 = wave_sum32(c1); c2 = wave_sum32(c2);
  k1 = wave_sum32(k1); k2 = wave_sum32(k2); w  = wave_sum32(w);

  __shared__ float sh[8][6];
  const int wave = threadIdx.x >> 5, lane = threadIdx.x & 31;
  if (lane == 0) {
    sh[wave][0] = c0; sh[wave][1] = c1; sh[wave][2] = c2;
    sh[wave][3] = k1; sh[wave][4] = k2; sh[wave][5] = w;
  }
  __syncthreads();
  if (threadIdx.x == 0) {
    float C0 = 0.f, C1 = 0.f, C2 = 0.f, K1 = 0.f, K2 = 0.f, W = 0.f;
#pragma unroll
    for (int i = 0; i < 8; ++i) {
      C0 += sh[i][0]; C1 += sh[i][1]; C2 += sh[i][2];
      K1 += sh[i][3]; K2 += sh[i][4]; W  += sh[i][5];
    }
    const float invBN = 1.0f / ((float)batch * (float)n_items);
    const float bce_m = -C0 * invBN;   // recon_x
    const float bce_r = -C1 * invBN;   // logits_rec
    const float bce_t = -C2 * invBN;   // logits_text
    const float BCE = (bce_m + bce_t + bce_r) * (1.0f / 3.0f);
    const float invBE = 1.0f / ((float)batch * (float)EMB_C);
    const float KLD1 = -0.5f * K1 * invBE;
    const float KLD2 = -0.5f * K2 * invBE;
    const float wass = W / (float)batch;
    out[0] = BCE + 0.5f * (KLD1 + KLD2) + wass;  // ANNEAL=1, EPSILON=1
    out[1] = BCE;
    out[2] = wass;
    out[3] = bce_r;
    out[4] = bce_t;
    out[5] = bce_m;
  }
}

extern "C" void kernel_launch(void* const* d_in, const int* in_sizes, int n_in,
                              void* d_out, int out_size, void* d_ws, size_t ws_size,
                              hipStream_t stream) {
  const float* recon = (const float*)d_in[0];
  const float* lrec  = (const float*)d_in[1];
  const float* ltext = (const float*)d_in[2];
  const float* x     = (const float*)d_in[3];
  const float* mu    = (const float*)d_in[4];
  const float* lv    = (const float*)d_in[5];
  const float* pmu   = (const float*)d_in[6];
  const float* plv   = (const float*)d_in[7];
  float* out = (float*)d_out;

  const int batch   = in_sizes[4] / EMB_C;        // 4096
  const int n_items = in_sizes[0] / batch;        // 20000
  const int emb_tot = in_sizes[4];                // batch * 256

  float* wsA = (float*)d_ws;                      // batch*3 per-row BCE contribs
  float* wsB = wsA + (size_t)batch * 3;           // 64*3 KLD/Wass partials
  const int NWG2 = 64;

  bce_rows_kernel<<<batch, 256, 0, stream>>>(
      (const float4*)recon, (const float4*)lrec, (const float4*)ltext,
      (const float4*)x, wsA, n_items / 4);
  kld_wass_kernel<<<NWG2, 256, 0, stream>>>(mu, lv, pmu, plv, wsB, emb_tot);
  finalize_kernel<<<1, 256, 0, stream>>>(wsA, wsB, batch, n_items, NWG2, out);
}